// RegionEncoder_60060822667896
// MI455X (gfx1250) — compile-verified
//
#include <hip/hip_runtime.h>

// RegionEncoder gather-multiply-max for MI455X (gfx1250, wave32).
// Pure bandwidth-bound gather: ~150 MB moved, ~55 MFLOP -> no WMMA applicable.
// One wave per (batch,pos): 32 lanes x float4 == one 512B embedding row.

typedef float v4f __attribute__((ext_vector_type(4)));

#define VOCAB   50000
#define EMB     128
#define RADIUS  3
#define REGION  7
#define BATCH   32
#define SEQ     1024
#define NPOS    (BATCH * SEQ)
#define WAVES_PER_BLOCK 8
#define THREADS (WAVES_PER_BLOCK * 32)

__global__ __launch_bounds__(THREADS)
void region_encoder_kernel(const int*   __restrict__ seq,  // [BATCH, SEQ]
                           const float* __restrict__ W,    // [VOCAB-1, EMB]
                           const float* __restrict__ U,    // [(VOCAB-1)*REGION, EMB]
                           float*       __restrict__ out)  // [BATCH, SEQ, EMB]
{
    const int lane = threadIdx.x & 31;
    // Make wave id explicitly uniform so index math + seq reads go scalar (SMEM).
    const int wave = __builtin_amdgcn_readfirstlane((int)(threadIdx.x >> 5));
    const int pos  = blockIdx.x * WAVES_PER_BLOCK + wave;   // grid covers NPOS exactly
    const int b    = pos >> 10;          // SEQ == 1024
    const int t    = pos & (SEQ - 1);

    const int* seqrow = seq + (long)b * SEQ;

    // Window tokens (wave-uniform). Clamp for safe always-load, track validity.
    int  tok[REGION];
    bool ok[REGION];
#pragma unroll
    for (int r = 0; r < REGION; ++r) {
        int s  = t - RADIUS + r;
        int sc = s < 0 ? 0 : (s > SEQ - 1 ? SEQ - 1 : s);
        int tk = seqrow[sc];
        ok[r]  = (s == sc) && (tk != 0);   // out-of-range or token 0 -> zero row
        tok[r] = tk;
    }
    const int tokc = tok[RADIUS];          // windows[t][RADIUS] == seq[t]

    const int e = lane * 4;                // this lane's 4 embedding floats

    // Center embedding. tokc==0 means W_full row 0 (zeros) -> mask at the end.
    const long wrow = (long)(tokc > 0 ? tokc - 1 : 0) * EMB;
    const v4f w = *(const v4f*)(W + wrow + e);

    // max over 7 slots; invalid slots contribute exact 0.0f.
    const float NEG_INF = -__builtin_huge_valf();
    v4f acc = { NEG_INF, NEG_INF, NEG_INF, NEG_INF };

#pragma unroll
    for (int r = 0; r < REGION; ++r) {
        const long urow = ((long)(ok[r] ? tok[r] - 1 : 0) * REGION + r) * (long)EMB;
        const v4f u = *(const v4f*)(U + urow + e);
        const v4f p = u * w;
        acc.x = fmaxf(acc.x, ok[r] ? p.x : 0.0f);
        acc.y = fmaxf(acc.y, ok[r] ? p.y : 0.0f);
        acc.z = fmaxf(acc.z, ok[r] ? p.z : 0.0f);
        acc.w = fmaxf(acc.w, ok[r] ? p.w : 0.0f);
    }

    if (tokc == 0) {                       // W_full[0] is the zero row
        acc = (v4f){0.0f, 0.0f, 0.0f, 0.0f};
    }

    // Streaming output (written once, never re-read this dispatch): NT store so
    // the 16.8MB result doesn't evict the U table from the 192MB L2.
    __builtin_nontemporal_store(acc, (v4f*)(out + (long)pos * EMB + e));
}

extern "C" void kernel_launch(void* const* d_in, const int* in_sizes, int n_in,
                              void* d_out, int out_size, void* d_ws, size_t ws_size,
                              hipStream_t stream) {
    (void)in_sizes; (void)n_in; (void)d_ws; (void)ws_size; (void)out_size;
    const int*   seq = (const int*)  d_in[0];
    const float* W   = (const float*)d_in[1];
    const float* U   = (const float*)d_in[2];
    float*       out = (float*)d_out;

    const int grid = NPOS / WAVES_PER_BLOCK;   // 4096 blocks x 8 waves = 32768 positions
    region_encoder_kernel<<<grid, THREADS, 0, stream>>>(seq, W, U, out);
}